// SplatCondenseNet_80229989089748
// MI455X (gfx1250) — compile-verified
//
#include <hip/hip_runtime.h>
#include <stdint.h>

typedef float v2f  __attribute__((ext_vector_type(2)));
typedef float v4f  __attribute__((ext_vector_type(4)));
typedef float v8f  __attribute__((ext_vector_type(8)));
typedef unsigned int u32x4 __attribute__((ext_vector_type(4)));
typedef int   i32x4 __attribute__((ext_vector_type(4)));
typedef int   i32x8 __attribute__((ext_vector_type(8)));

#define N_LEAVES 524288
#define D_FEAT   128
#define FSTRIDE  132          // padded LDS row stride (floats): +4 mod 64 banks -> conflict-free
#define FSTRIDE4 (FSTRIDE / 4)
#define LPB      64           // leaves per block
#define NBLOCKS  (N_LEAVES / LPB)   // 8192

// output row offsets (rows of 128 floats), layers 1..6 concatenated
#define O_L1 0
#define O_L2 262144
#define O_L3 393216
#define O_L4 458752
#define O_L5 491520
#define O_L6 507904

__device__ __forceinline__ int src_row_of_group(int g) {
  // groups 0..7: layer3 sources = lx rows 8g ; 8..11: layer4 = l1 rows (feat 64+8(g-8))
  // 12..13: layer5 = l2 rows (feat 96+8(g-12))
  return (g < 8) ? 8 * g : (g < 12) ? 64 + 8 * (g - 8) : 96 + 8 * (g - 12);
}

__global__ __launch_bounds__(256)
void splat_condense_fused(const float* __restrict__ x0,
                          const float* __restrict__ w_attn,
                          float* __restrict__ out) {
  // rows 0-63: lx tile, 64-95: layer1, 96-111: layer2
  __shared__ float feat[112][FSTRIDE];
  __shared__ float l3buf[8][D_FEAT];
  __shared__ float lw[D_FEAT];
  __shared__ float sc[112];
  __shared__ float wts[15][8];

  v4f (*feat4)[FSTRIDE4] = (v4f (*)[FSTRIDE4])&feat[0][0];
  v4f (*l3buf4)[D_FEAT / 4] = (v4f (*)[D_FEAT / 4])&l3buf[0][0];

  const int tid  = threadIdx.x;
  const int lane = tid & 31;
  const int wave = __builtin_amdgcn_readfirstlane((int)tid) >> 5;  // uniform per wave
  const int blk  = blockIdx.x;

  // ---------------- Stage 0: TDM async DMA of this block's 64x128 f32 tile ----------------
  // Each wave DMAs its own 8x128 slice (per-wave issue: TDM ignores EXEC, so never rely on
  // predication to suppress it). pad: +4 DWORDs per 128 DWORDs -> LDS row stride = 132 floats.
  {
    uint32_t ldsOff = (uint32_t)(uintptr_t)&feat[wave * 8][0];     // AS3 offset = low 32 bits
    uint64_t ga = (uint64_t)(uintptr_t)(x0 + ((size_t)blk * LPB + (size_t)wave * 8) * D_FEAT);
    u32x4 g0;
    g0[0] = 1u;                                   // count=1, user mode
    g0[1] = ldsOff;                               // lds_addr (bytes)
    g0[2] = (uint32_t)ga;                         // global_addr[31:0]
    g0[3] = (uint32_t)(ga >> 32) | (2u << 30);    // global_addr[56:32] | type=2
    i32x8 g1;
    g1[0] = (int)((2u << 16)                      // data_size = 4 bytes
                | (1u << 20)                      // pad_enable
                | (6u << 22)                      // pad_interval: 128 DWORDs
                | (3u << 25));                    // pad_amount:   4 DWORDs
    g1[1] = (int)(128u << 16);                    // tensor_dim0 = 128 elements
    g1[2] = 0;                                    // tensor_dim1 lo16 (524288 & 0xffff = 0)
    g1[3] = (int)(8u | (128u << 16));             // tensor_dim1 hi16 = 8 ; tile_dim0 = 128
    g1[4] = 8;                                    // tile_dim1 = 8 rows ; tile_dim2 = 0 (2D)
    g1[5] = 128;                                  // tensor_dim0_stride = 128 elements
    g1[6] = 0;
    g1[7] = 0;
    i32x4 z4 = {0, 0, 0, 0};
#if __clang_major__ >= 23
    i32x8 z8 = {0, 0, 0, 0, 0, 0, 0, 0};
    __builtin_amdgcn_tensor_load_to_lds(g0, g1, z4, z4, z8, 0);
#else
    __builtin_amdgcn_tensor_load_to_lds(g0, g1, z4, z4, 0);
#endif
  }
  if (tid < 32) {                                  // one b128 load per lane: whole w vector
    ((v4f*)lw)[tid] = ((const v4f*)w_attn)[tid];
  }
  __builtin_amdgcn_s_wait_tensorcnt(0);
  __syncthreads();

  // ---------------- Stage 1: layer1 mean pool (32 rows, float4) ----------------
  for (int idx = tid; idx < 32 * 32; idx += 256) {
    int r = idx >> 5, q = idx & 31;
    v4f v = 0.5f * (feat4[2 * r][q] + feat4[2 * r + 1][q]);
    feat4[64 + r][q] = v;
    __builtin_nontemporal_store(v, (v4f*)&out[(size_t)(O_L1 + blk * 32 + r) * D_FEAT + 4 * q]);
  }
  __syncthreads();

  // ---------------- Stage 2: layer2 mean pool (16 rows, float4) ----------------
  for (int idx = tid; idx < 16 * 32; idx += 256) {
    int r = idx >> 5, q = idx & 31;
    v4f v = 0.5f * (feat4[64 + 2 * r][q] + feat4[64 + 2 * r + 1][q]);
    feat4[96 + r][q] = v;
    __builtin_nontemporal_store(v, (v4f*)&out[(size_t)(O_L2 + blk * 16 + r) * D_FEAT + 4 * q]);
  }
  __syncthreads();

  // ---------------- Stage 3: attention scores for feat rows 0..111 via WMMA f32 16x16x4 ----
  // A = 16 source rows x K4, B = w_attn in column 0 (other columns zero), accumulate K=128.
  // D column 0 = dot(row, w). Wave t owns 16-row tile t (7 tiles cover rows 0..111).
  const float scale = 0.08838834764831845f;       // 128^-0.5
  if (wave < 7) {
    const int half = lane >> 4, m = lane & 15;
    const float* rp = &feat[wave * 16 + m][0];
    const bool col0 = (m == 0);
    v8f c = {0.f, 0.f, 0.f, 0.f, 0.f, 0.f, 0.f, 0.f};
#pragma unroll
    for (int kk = 0; kk < 32; ++kk) {
      const int col = kk * 4 + half * 2;
      v2f a  = *(const v2f*)&rp[col];
      v2f bv = *(const v2f*)&lw[col];            // unconditional load (one ds_load_b64)
      v2f b;
      b[0] = col0 ? bv[0] : 0.0f;                // v_cndmask, no EXEC branching
      b[1] = col0 ? bv[1] : 0.0f;
      c = __builtin_amdgcn_wmma_f32_16x16x4_f32(false, a, false, b, (short)0, c, false, false);
    }
    if (col0) {   // lanes 0 (rows 0-7 of tile) and 16 (rows 8-15) hold column 0
#pragma unroll
      for (int v = 0; v < 8; ++v) sc[wave * 16 + half * 8 + v] = c[v] * scale;
    }
  }
  __syncthreads();

  // ---------------- Stage 4: softmax weights for groups 0..13 ----------------
  if (tid < 14) {
    const int g = tid, sr = src_row_of_group(g);
    float s[8], mx = -3.402823466e38f;
#pragma unroll
    for (int k = 0; k < 8; ++k) { s[k] = sc[sr + k]; mx = fmaxf(mx, s[k]); }
    float sum = 0.f;
#pragma unroll
    for (int k = 0; k < 8; ++k) { s[k] = __expf(s[k] - mx); sum += s[k]; }
    const float inv = 1.0f / sum;
#pragma unroll
    for (int k = 0; k < 8; ++k) wts[g][k] = s[k] * inv;
  }
  __syncthreads();

  // ---------------- Stage 5: weighted sums for layers 3,4,5 (float4) ----------------
  for (int idx = tid; idx < 14 * 32; idx += 256) {
    int g = idx >> 5, q = idx & 31;
    const int sr = src_row_of_group(g);
    v4f acc = {0.f, 0.f, 0.f, 0.f};
#pragma unroll
    for (int k = 0; k < 8; ++k) acc += wts[g][k] * feat4[sr + k][q];
    size_t orow;
    if (g < 8)       orow = (size_t)O_L3 + (size_t)blk * 8 + g;
    else if (g < 12) orow = (size_t)O_L4 + (size_t)blk * 4 + (g - 8);
    else             orow = (size_t)O_L5 + (size_t)blk * 2 + (g - 12);
    __builtin_nontemporal_store(acc, (v4f*)&out[orow * D_FEAT + 4 * q]);
    if (g < 8) l3buf4[g][q] = acc;                // layer3 feeds layer6
  }
  __syncthreads();

  // ---------------- Stage 6: layer6 scores over the 8 layer3 rows (wave 0, WMMA) --------
  if (wave == 0) {
    const int half = lane >> 4, m = lane & 15;
    const float* rp = &l3buf[(m < 8) ? m : 7][0];  // clamp padding rows (results unused)
    const bool col0 = (m == 0);
    v8f c = {0.f, 0.f, 0.f, 0.f, 0.f, 0.f, 0.f, 0.f};
#pragma unroll
    for (int kk = 0; kk < 32; ++kk) {
      const int col = kk * 4 + half * 2;
      v2f a  = *(const v2f*)&rp[col];
      v2f bv = *(const v2f*)&lw[col];
      v2f b;
      b[0] = col0 ? bv[0] : 0.0f;
      b[1] = col0 ? bv[1] : 0.0f;
      c = __builtin_amdgcn_wmma_f32_16x16x4_f32(false, a, false, b, (short)0, c, false, false);
    }
    if (col0) {
#pragma unroll
      for (int v = 0; v < 8; ++v) sc[half * 8 + v] = c[v] * scale;   // only sc[0..7] used
    }
  }
  __syncthreads();

  if (tid == 0) {
    float s[8], mx = -3.402823466e38f;
#pragma unroll
    for (int k = 0; k < 8; ++k) { s[k] = sc[k]; mx = fmaxf(mx, s[k]); }
    float sum = 0.f;
#pragma unroll
    for (int k = 0; k < 8; ++k) { s[k] = __expf(s[k] - mx); sum += s[k]; }
    const float inv = 1.0f / sum;
#pragma unroll
    for (int k = 0; k < 8; ++k) wts[14][k] = s[k] * inv;
  }
  __syncthreads();

  // ---------------- Stage 7: layer6 pooled row (float4) ----------------
  if (tid < 32) {
    v4f acc = {0.f, 0.f, 0.f, 0.f};
#pragma unroll
    for (int k = 0; k < 8; ++k) acc += wts[14][k] * l3buf4[k][tid];
    __builtin_nontemporal_store(acc, (v4f*)&out[((size_t)O_L6 + blk) * D_FEAT + 4 * tid]);
  }
}

extern "C" void kernel_launch(void* const* d_in, const int* in_sizes, int n_in,
                              void* d_out, int out_size, void* d_ws, size_t ws_size,
                              hipStream_t stream) {
  (void)in_sizes; (void)n_in; (void)out_size; (void)d_ws; (void)ws_size;
  const float* x0 = (const float*)d_in[0];
  const float* w  = (const float*)d_in[1];
  float* out      = (float*)d_out;
  splat_condense_fused<<<NBLOCKS, 256, 0, stream>>>(x0, w, out);
}